// GRNN_90013924590102
// MI455X (gfx1250) — compile-verified
//
#include <hip/hip_runtime.h>

#define DIMX   20
#define NNODES 200000
#define NEDGES 6400000
#define ITERS  3

typedef float v2f __attribute__((ext_vector_type(2)));
typedef float v8f __attribute__((ext_vector_type(8)));

// One 16x20 (M x K) tile times one 20x20 weight (stored K-major, padded to
// [20][32] with zero columns) -> two 16x16 f32 accumulators (N-tiles 0..15 and
// 16..31, second half zero-padded). Uses V_WMMA_F32_16X16X4_F32, K in 5 steps.
// Fragment layout (ISA 7.12.2): A/B vgpr j, half hi -> K = kb*4 + 2*hi + j,
// M/N = lane&15. C/D vgpr q -> M = q + 8*hi, N = lane&15.
__device__ __forceinline__ void wave_mm20(const float* __restrict__ tile, // [16][20]
                                          const float* __restrict__ wt,   // [20][32]
                                          v8f& c0, v8f& c1,
                                          const int lo, const int hi) {
#pragma unroll
  for (int kb = 0; kb < 5; ++kb) {
    const int k0 = kb * 4 + 2 * hi;
    v2f a, b0, b1;
    a[0]  = tile[lo * DIMX + k0];
    a[1]  = tile[lo * DIMX + k0 + 1];
    b0[0] = wt[k0 * 32 + lo];
    b0[1] = wt[(k0 + 1) * 32 + lo];
    b1[0] = wt[k0 * 32 + 16 + lo];        // zero-padded cols >= 20
    b1[1] = wt[(k0 + 1) * 32 + 16 + lo];
    c0 = __builtin_amdgcn_wmma_f32_16x16x4_f32(false, a, false, b0, (short)0, c0, false, false);
    c1 = __builtin_amdgcn_wmma_f32_16x16x4_f32(false, a, false, b1, (short)0, c1, false, false);
  }
}

__global__ __launch_bounds__(256) void zero_kernel(float* __restrict__ p, int n) {
  int i = blockIdx.x * 256 + threadIdx.x;
  if (i < n) p[i] = 0.0f;
}

// One edge per wave32; lanes 0..19 move one f32 each (coalesced 80B row
// gather + 80B row of L2 float atomics). Inactive src OR dst => skip: the
// reference drops h_new for inactive dst and masks msg by af[src], so this is
// exactly equivalent work elimination.
__global__ __launch_bounds__(256) void edge_scatter_kernel(
    const float* __restrict__ h, const int* __restrict__ depth,
    const int* __restrict__ src, const int* __restrict__ dst,
    float* __restrict__ x, const int thresh) {
  const long long gid = (long long)blockIdx.x * 256 + threadIdx.x;
  const long long e = gid >> 5;
  const int lane = (int)(gid & 31);
  if (e >= (long long)NEDGES) return;
  const int s = src[e];
  const int d = dst[e];
  if (depth[s] > thresh || depth[d] > thresh) return;
  if (lane < DIMX)
    atomicAdd(&x[(long long)d * DIMX + lane], h[(long long)s * DIMX + lane]);
}

__global__ __launch_bounds__(256) void node_update_kernel(
    const float* __restrict__ x, float* __restrict__ h,
    const int* __restrict__ depth,
    const float* __restrict__ Wz_w, const float* __restrict__ Wz_b,
    const float* __restrict__ Uz_w, const float* __restrict__ Uz_b,
    const float* __restrict__ Wr_w, const float* __restrict__ Wr_b,
    const float* __restrict__ Ur_w, const float* __restrict__ Ur_b,
    const float* __restrict__ Wh_w, const float* __restrict__ Wh_b,
    const float* __restrict__ Uh_w, const float* __restrict__ Uh_b,
    const int thresh) {
  __shared__ float sWT[6][20][32];   // W^T, zero-padded columns 20..31
  __shared__ float sB[6][32];        // biases, zero-padded
  __shared__ float sX[8][16][DIMX];  // per-wave X tile
  __shared__ float sH[8][16][DIMX];  // per-wave H tile
  __shared__ float sT[8][16][DIMX];  // per-wave r*h tile (C-layout -> A-layout)

  const int tid  = threadIdx.x;
  const int w    = tid >> 5;
  const int lane = tid & 31;
  const int lo   = lane & 15;
  const int hi   = lane >> 4;

  const float* wptr[6] = {Wz_w, Uz_w, Wr_w, Ur_w, Wh_w, Uh_w};
  const float* bptr[6] = {Wz_b, Uz_b, Wr_b, Ur_b, Wh_b, Uh_b};
#pragma unroll
  for (int m = 0; m < 6; ++m) {
    for (int i = tid; i < 640; i += 256) {
      const int k = i >> 5, n = i & 31;
      // torch-style weight is [out,in]; store transposed [in(k)][out(n)]
      sWT[m][k][n] = (n < DIMX) ? wptr[m][n * DIMX + k] : 0.0f;
    }
    if (tid < 32) sB[m][tid] = (tid < DIMX) ? bptr[m][tid] : 0.0f;
  }

  const int group    = blockIdx.x * 8 + w;      // 16-node group (N % 16 == 0)
  const int nodeBase = group * 16;
  const bool wvalid  = nodeBase < NNODES;

  if (wvalid) {
    for (int i = lane; i < 16 * DIMX; i += 32) {
      sX[w][i / DIMX][i % DIMX] = x[(long long)nodeBase * DIMX + i];
      sH[w][i / DIMX][i % DIMX] = h[(long long)nodeBase * DIMX + i];
    }
  }
  __syncthreads();

  v8f z0 = {0.f,0.f,0.f,0.f,0.f,0.f,0.f,0.f};
  v8f z1 = {0.f,0.f,0.f,0.f,0.f,0.f,0.f,0.f};

  if (wvalid) {
    v8f cz0 = {0.f,0.f,0.f,0.f,0.f,0.f,0.f,0.f};
    v8f cz1 = {0.f,0.f,0.f,0.f,0.f,0.f,0.f,0.f};
    wave_mm20(&sX[w][0][0], &sWT[0][0][0], cz0, cz1, lo, hi);  // X @ Wz^T
    wave_mm20(&sH[w][0][0], &sWT[1][0][0], cz0, cz1, lo, hi);  // + H @ Uz^T
    v8f cr0 = {0.f,0.f,0.f,0.f,0.f,0.f,0.f,0.f};
    v8f cr1 = {0.f,0.f,0.f,0.f,0.f,0.f,0.f,0.f};
    wave_mm20(&sX[w][0][0], &sWT[2][0][0], cr0, cr1, lo, hi);  // X @ Wr^T
    wave_mm20(&sH[w][0][0], &sWT[3][0][0], cr0, cr1, lo, hi);  // + H @ Ur^T

    const float bz_0 = sB[0][lo] + sB[1][lo];
    const float bz_1 = sB[0][16 + lo] + sB[1][16 + lo];
    const float br_0 = sB[2][lo] + sB[3][lo];
    const float br_1 = sB[2][16 + lo] + sB[3][16 + lo];
#pragma unroll
    for (int q = 0; q < 8; ++q) {
      const int m = q + 8 * hi;
      z0[q] = 1.0f / (1.0f + expf(-(cz0[q] + bz_0)));
      z1[q] = 1.0f / (1.0f + expf(-(cz1[q] + bz_1)));
      const float rv0 = 1.0f / (1.0f + expf(-(cr0[q] + br_0)));
      sT[w][m][lo] = rv0 * sH[w][m][lo];
      if (lo < 4) {
        const float rv1 = 1.0f / (1.0f + expf(-(cr1[q] + br_1)));
        sT[w][m][16 + lo] = rv1 * sH[w][m][16 + lo];
      }
    }
  }
  __syncthreads();   // publish r*h tile (C-layout writes -> A-layout reads)

  if (wvalid) {
    v8f ch0 = {0.f,0.f,0.f,0.f,0.f,0.f,0.f,0.f};
    v8f ch1 = {0.f,0.f,0.f,0.f,0.f,0.f,0.f,0.f};
    wave_mm20(&sX[w][0][0], &sWT[4][0][0], ch0, ch1, lo, hi);  // X @ Wh^T
    wave_mm20(&sT[w][0][0], &sWT[5][0][0], ch0, ch1, lo, hi);  // + (r*h) @ Uh^T
    const float bh_0 = sB[4][lo] + sB[5][lo];
    const float bh_1 = sB[4][16 + lo] + sB[5][16 + lo];
#pragma unroll
    for (int q = 0; q < 8; ++q) {
      const int m = q + 8 * hi;
      const int node = nodeBase + m;
      const bool act = depth[node] <= thresh;
      const float hold0 = sH[w][m][lo];
      const float hc0 = tanhf(ch0[q] + bh_0);
      const float hn0 = z0[q] * hold0 + (1.0f - z0[q]) * hc0;
      if (act) h[(long long)node * DIMX + lo] = hn0;
      if (lo < 4) {
        const float hold1 = sH[w][m][16 + lo];
        const float hc1 = tanhf(ch1[q] + bh_1);
        const float hn1 = z1[q] * hold1 + (1.0f - z1[q]) * hc1;
        if (act) h[(long long)node * DIMX + 16 + lo] = hn1;
      }
    }
  }
}

__global__ __launch_bounds__(256) void finalize_kernel(
    const float* __restrict__ h, const int* __restrict__ depth,
    float* __restrict__ out) {
  const long long i = (long long)blockIdx.x * 256 + threadIdx.x;
  if (i >= (long long)NNODES * DIMX) return;
  const int node = (int)(i / DIMX);
  out[i] = (depth[node] <= (3 - (ITERS - 1))) ? h[i] : 0.0f;
}

extern "C" void kernel_launch(void* const* d_in, const int* in_sizes, int n_in,
                              void* d_out, int out_size, void* d_ws, size_t ws_size,
                              hipStream_t stream) {
  const float* h_in  = (const float*)d_in[0];
  const int*   depth = (const int*)d_in[1];
  const int*   esrc  = (const int*)d_in[2];
  const int*   edst  = (const int*)d_in[3];
  const float* Wz_w = (const float*)d_in[4];  const float* Wz_b = (const float*)d_in[5];
  const float* Uz_w = (const float*)d_in[6];  const float* Uz_b = (const float*)d_in[7];
  const float* Wr_w = (const float*)d_in[8];  const float* Wr_b = (const float*)d_in[9];
  const float* Ur_w = (const float*)d_in[10]; const float* Ur_b = (const float*)d_in[11];
  const float* Wh_w = (const float*)d_in[12]; const float* Wh_b = (const float*)d_in[13];
  const float* Uh_w = (const float*)d_in[14]; const float* Uh_b = (const float*)d_in[15];

  float* hbuf = (float*)d_ws;                       // N*20 f32 (16 MB)
  float* xbuf = hbuf + (size_t)NNODES * DIMX;       // N*20 f32 (16 MB)

  hipMemcpyAsync(hbuf, h_in, sizeof(float) * (size_t)NNODES * DIMX,
                 hipMemcpyDeviceToDevice, stream);

  const int  nElem         = NNODES * DIMX;
  const int  zeroBlocks    = (nElem + 255) / 256;
  const long long sthreads = (long long)NEDGES * 32;
  const int  scatterBlocks = (int)((sthreads + 255) / 256);
  const int  updateBlocks  = (NNODES / 16 + 7) / 8;

  for (int it = 0; it < ITERS; ++it) {
    const int thresh = 3 - it;   // active: depth + it <= 3
    zero_kernel<<<zeroBlocks, 256, 0, stream>>>(xbuf, nElem);
    edge_scatter_kernel<<<scatterBlocks, 256, 0, stream>>>(hbuf, depth, esrc, edst,
                                                           xbuf, thresh);
    node_update_kernel<<<updateBlocks, 256, 0, stream>>>(xbuf, hbuf, depth,
        Wz_w, Wz_b, Uz_w, Uz_b, Wr_w, Wr_b, Ur_w, Ur_b, Wh_w, Wh_b, Uh_w, Uh_b,
        thresh);
  }
  finalize_kernel<<<zeroBlocks, 256, 0, stream>>>(hbuf, depth, (float*)d_out);
}